// MultiSamplePatches_2156073583008
// MI455X (gfx1250) — compile-verified
//
#include <hip/hip_runtime.h>
#include <hip/hip_bf16.h>
#include <stdint.h>

#define B_      16
#define C_      3
#define NP      16
#define PATCHSZ 64
#define T_      20480
#define PATCH_ELEMS (PATCHSZ * PATCHSZ * C_)   // 12288 floats = 48 KB

// --- TDM builtin availability / arity detection -----------------------------
#if __has_include(<hip/amd_detail/amd_gfx1250_TDM.h>)
#define MSP_TDM_6ARG 1        // amdgpu-toolchain (clang-23): 6-arg builtin
#endif
#if defined(__has_builtin)
#if __has_builtin(__builtin_amdgcn_tensor_load_to_lds)
#define MSP_HAS_TDM 1
#endif
#endif

typedef unsigned int msp_v4u __attribute__((ext_vector_type(4)));
typedef int          msp_v4i __attribute__((ext_vector_type(4)));
typedef int          msp_v8i __attribute__((ext_vector_type(8)));

// ---------------------------------------------------------------------------
// Kernel A: per-batch inclusive prefix sum (CDF) of attention[B, T]
// ---------------------------------------------------------------------------
__global__ void msp_scan_kernel(const float* __restrict__ att,
                                float* __restrict__ prefix) {
    __shared__ float cs[1024];
    const int b = blockIdx.x;
    const int t = threadIdx.x;
    const int CH = T_ / 1024;                 // 20
    const int base = b * T_ + t * CH;

    float loc[CH];
    float s = 0.f;
#pragma unroll
    for (int k = 0; k < CH; ++k) { loc[k] = att[base + k]; s += loc[k]; }
    cs[t] = s;
    __syncthreads();

    for (int off = 1; off < 1024; off <<= 1) {
        float v = (t >= off) ? cs[t - off] : 0.f;
        __syncthreads();
        cs[t] += v;
        __syncthreads();
    }

    float run = cs[t] - s;
#pragma unroll
    for (int k = 0; k < CH; ++k) { run += loc[k]; prefix[base + k] = run; }
}

// ---------------------------------------------------------------------------
// Kernel B: inverse-CDF multinomial sampling + coordinate mapping.
// ---------------------------------------------------------------------------
__device__ __forceinline__ uint32_t msp_pcg(uint32_t x) {
    x = x * 747796405u + 2891336453u;
    uint32_t w = ((x >> ((x >> 28u) + 4u)) ^ x) * 277803737u;
    return (w >> 22u) ^ w;
}

__global__ void msp_sample_kernel(const float* __restrict__ att,
                                  const float* __restrict__ prefix,
                                  const int*   __restrict__ map_index,
                                  float* __restrict__ out_att,
                                  int4*  __restrict__ samples) {
    const int i = blockIdx.x * blockDim.x + threadIdx.x;
    if (i >= B_ * NP) return;
    const int b = i / NP;
    const int n = i - b * NP;

    const float* p = prefix + (size_t)b * T_;
    const float total = p[T_ - 1];

    uint32_t h = msp_pcg(0x9e3779b9u ^ (uint32_t)(b * 65537 + n * 257));
    float u = (float)(h >> 8) * (1.0f / 16777216.0f);
    if (u > 0.99999994f) u = 0.99999994f;
    const float target = u * total;

    int lo = 0, hi = T_ - 1;
    while (lo < hi) {
        int mid = (lo + hi) >> 1;
        if (p[mid] < target) lo = mid + 1; else hi = mid;
    }
    const int idx = lo;

    out_att[i] = att[(size_t)b * T_ + idx];

    const int s   = map_index[idx * 3 + 0];
    const int row = map_index[idx * 3 + 1];
    const int col = map_index[idx * 3 + 2];
    const float low   = (s == 0) ? 128.f  : 64.f;
    const float hires = (s == 0) ? 1024.f : 512.f;

    const float cy = ((float)row + 0.5f) / low * hires;
    const float cx = ((float)col + 0.5f) / low * hires;
    const float mx = hires - (float)PATCHSZ;
    float syf = rintf(cy - (float)PATCHSZ * 0.5f);
    float sxf = rintf(cx - (float)PATCHSZ * 0.5f);
    syf = fminf(fmaxf(syf, 0.f), mx);
    sxf = fminf(fmaxf(sxf, 0.f), mx);

    samples[i] = make_int4(s, (int)syf, (int)sxf, idx);
}

// ---------------------------------------------------------------------------
// Kernel C: patch gather. One block per (b, n); 256 threads.
// TDM: one TENSOR_LOAD_TO_LDS per channel DMAs a 64x64 strided tile into LDS
// (CHW order), then all waves drain to coalesced HWC global stores.
// ---------------------------------------------------------------------------
__global__ void msp_gather_kernel(const float* __restrict__ xh0,  // [B,3,1024,1024]
                                  const float* __restrict__ xh1,  // [B,3,512,512]
                                  const int4*  __restrict__ samples,
                                  float* __restrict__ out) {
    __shared__ float lds[PATCH_ELEMS];         // 48 KB, CHW order
    const int blk = blockIdx.x;                // b * NP + n
    const int b   = blk / NP;
    const int tid = threadIdx.x;

    const int4 sm = samples[blk];
    // Make the descriptor fields provably uniform (SGPR-resident).
    const int s  = __builtin_amdgcn_readfirstlane(sm.x);
    const int sy = __builtin_amdgcn_readfirstlane(sm.y);
    const int sx = __builtin_amdgcn_readfirstlane(sm.z);
    const float* base = s ? xh1 : xh0;
    const int H = s ? 512 : 1024;              // square images, row pitch == H

#if defined(MSP_HAS_TDM)
    if (tid < 32) {                            // wave 0 issues the DMAs
        const uint32_t dimu = (uint32_t)H;
#pragma unroll
        for (int c = 0; c < C_; ++c) {
            const uint64_t gaddr = (uint64_t)(uintptr_t)
                (base + ((size_t)(b * C_ + c) * H + (size_t)sy) * (size_t)H
                      + (size_t)sx);
            const uint32_t laddr = (uint32_t)(uintptr_t)(&lds[c << 12]);

            // ---- D# group 0 (128b): count=1 | lds_addr | global_addr | type=2
            msp_v4u g0;
            g0.x = 1u;                                         // count=1, user desc
            g0.y = laddr;                                      // lds_addr
            g0.z = (uint32_t)gaddr;                            // global_addr[31:0]
            g0.w = (uint32_t)((gaddr >> 32) & 0x01FFFFFFu)     // global_addr[56:32]
                 | (2u << 30);                                 // type = 2 (image)

            // ---- D# group 1 (256b)
            msp_v8i g1;
            g1[0] = 0x20000;                                   // data_size=2 -> 4B
            g1[1] = (int)((dimu & 0xFFFFu) << 16);             // tensor_dim0[15:0]
            g1[2] = (int)(((dimu >> 16) & 0xFFFFu)             // tensor_dim0[31:16]
                    | ((dimu & 0xFFFFu) << 16));               // tensor_dim1[15:0]
            g1[3] = (int)(((dimu >> 16) & 0xFFFFu)             // tensor_dim1[31:16]
                    | ((uint32_t)PATCHSZ << 16));              // tile_dim0 = 64
            g1[4] = (int)(uint32_t)PATCHSZ;                    // tile_dim1=64, tile_dim2=0
            g1[5] = (int)dimu;                                 // tensor_dim0_stride[31:0] = W
            g1[6] = 0;                                         // stride0[47:32]=0, stride1[15:0]=0
            g1[7] = 0;                                         // stride1[47:16]=0

            msp_v4i gz = {0, 0, 0, 0};
#if defined(MSP_TDM_6ARG)
            msp_v8i gz8 = {0, 0, 0, 0, 0, 0, 0, 0};
            __builtin_amdgcn_tensor_load_to_lds(g0, g1, gz, gz, gz8, 0);
#else
            __builtin_amdgcn_tensor_load_to_lds(g0, g1, gz, gz, 0);
#endif
        }
        __builtin_amdgcn_s_wait_tensorcnt(0);
    }
#else
    // Fallback: per-lane async global->LDS B32 copies.
#pragma unroll
    for (int it = 0; it < PATCH_ELEMS / 256; ++it) {
        const int e   = it * 256 + tid;
        const int c   = e >> 12;
        const int rem = e & 4095;
        const int ph  = rem >> 6;
        const int pw  = rem & 63;
        const uint64_t ga = (uint64_t)(uintptr_t)
            (base + ((size_t)(b * C_ + c) * H + (size_t)(sy + ph)) * (size_t)H
                  + (size_t)(sx + pw));
        const uint32_t laddr = (uint32_t)(uintptr_t)(&lds[e]);
        asm volatile("global_load_async_to_lds_b32 %0, %1, off"
                     :: "v"(laddr), "v"(ga) : "memory");
    }
    asm volatile("s_wait_asynccnt 0x0" ::: "memory");
#endif
    __syncthreads();

    // Drain: output element e within patch == (ph*64+pw)*3 + c, so iterating e
    // by thread id gives fully coalesced global stores.
    const size_t obase = (size_t)blk * PATCH_ELEMS;
#pragma unroll
    for (int e = tid; e < PATCH_ELEMS; e += 256) {
        const int c   = e % 3;
        const int pxy = e / 3;
        const int pw  = pxy & 63;
        const int ph  = pxy >> 6;
        out[obase + e] = lds[(c << 12) | (ph << 6) | pw];
    }
}

// ---------------------------------------------------------------------------
extern "C" void kernel_launch(void* const* d_in, const int* in_sizes, int n_in,
                              void* d_out, int out_size, void* d_ws, size_t ws_size,
                              hipStream_t stream) {
    // setup_inputs order: x_low0, x_low1, x_high0, x_high1, attention, map_index
    const float* x_high0   = (const float*)d_in[2];
    const float* x_high1   = (const float*)d_in[3];
    const float* attention = (const float*)d_in[4];
    const int*   map_index = (const int*)d_in[5];

    float* out = (float*)d_out;
    float* prefix  = (float*)d_ws;                                   // B*T floats
    int4*  samples = (int4*)((char*)d_ws + (size_t)B_ * T_ * sizeof(float));

    msp_scan_kernel<<<B_, 1024, 0, stream>>>(attention, prefix);

    float* out_att = out + (size_t)B_ * NP * PATCHSZ * PATCHSZ * C_; // 3,145,728
    msp_sample_kernel<<<1, 256, 0, stream>>>(attention, prefix, map_index,
                                             out_att, samples);

    msp_gather_kernel<<<B_ * NP, 256, 0, stream>>>(x_high0, x_high1, samples, out);
}